// KDEGaussian2_37915971289662
// MI455X (gfx1250) — compile-verified
//
#include <hip/hip_runtime.h>

// KDE Gaussian 2D: out[j] = sum_i mask_i * exp(-|s_i - l_j|^2 / BW^2), normalized.
// BW = 0.5 -> 1/BW^2 = 4. Work in log2 domain so v_exp_f32 (2^x) is the kernel.

#define LOG2E 1.44269504088896340736f
#define BW2_INV 4.0f

typedef __attribute__((ext_vector_type(2))) float v2f;
typedef __attribute__((ext_vector_type(8))) float v8f;

// ---------------------------------------------------------------------------
// Kernel 1: absmax over locations (single block)
// ---------------------------------------------------------------------------
__global__ void kde_absmax(const float* __restrict__ loc, float* __restrict__ absmax, int nl) {
    __shared__ float sx[256];
    __shared__ float sy[256];
    float mx = 0.f, my = 0.f;
    for (int i = threadIdx.x; i < nl; i += blockDim.x) {
        mx = fmaxf(mx, fabsf(loc[2 * i + 0]));
        my = fmaxf(my, fabsf(loc[2 * i + 1]));
    }
    sx[threadIdx.x] = mx;
    sy[threadIdx.x] = my;
    __syncthreads();
    for (int s = blockDim.x >> 1; s > 0; s >>= 1) {
        if ((int)threadIdx.x < s) {
            sx[threadIdx.x] = fmaxf(sx[threadIdx.x], sx[threadIdx.x + s]);
            sy[threadIdx.x] = fmaxf(sy[threadIdx.x], sy[threadIdx.x + s]);
        }
        __syncthreads();
    }
    if (threadIdx.x == 0) {
        absmax[0] = sx[0];
        absmax[1] = sy[0];
    }
}

// ---------------------------------------------------------------------------
// Kernel 2: prep samples into K=4 rows (log2 domain, mask folded in)
//   a = [2x*C, 2y*C, -(x^2+y^2)*C, C]  with C = log2e / BW^2
//   masked-out sample -> [0,0,-16384,0] : dot = -16384 -> exp2 = 0
// ---------------------------------------------------------------------------
__global__ void kde_prep_samples(const float* __restrict__ smp,
                                 const float* __restrict__ absmax,
                                 float4* __restrict__ A, int ns) {
    int i = blockIdx.x * blockDim.x + threadIdx.x;
    if (i >= ns) return;
    float x = smp[2 * i + 0];
    float y = smp[2 * i + 1];
    bool in = (fabsf(x) < absmax[0]) && (fabsf(y) < absmax[1]);
    const float C = LOG2E * BW2_INV;
    float4 a;
    if (in) a = make_float4(2.f * x * C, 2.f * y * C, -(x * x + y * y) * C, C);
    else    a = make_float4(0.f, 0.f, -16384.0f, 0.f);
    A[i] = a;
}

// ---------------------------------------------------------------------------
// Kernel 3: prep locations:  b = [x, y, 1, -(x^2+y^2)] ; also zero d_out
// ---------------------------------------------------------------------------
__global__ void kde_prep_locations(const float* __restrict__ loc,
                                   float4* __restrict__ B,
                                   float* __restrict__ out, int nl) {
    int i = blockIdx.x * blockDim.x + threadIdx.x;
    if (i >= nl) return;
    float x = loc[2 * i + 0];
    float y = loc[2 * i + 1];
    B[i] = make_float4(x, y, 1.f, -(x * x + y * y));
    out[i] = 0.f;
}

// ---------------------------------------------------------------------------
// Kernel 4: WMMA main loop.
//   Each wave32 owns a tile of 16 locations (N) and one sample chunk (M).
//   Per iteration: V_WMMA_F32_16X16X4_F32 -> 256 dot products (log2 domain),
//   then 8x v_exp_f32 + 8x v_add_f32 per lane. Accumulate over the chunk,
//   reduce over M (8 VGPRs + cross-half shuffle), atomic add into out.
//
//   A-operand layout (16x4 f32, 2 VGPRs): lanes 0-15 -> K0,K1 of row m;
//   lanes 16-31 -> K2,K3 of row m.  B-operand (4x16) mirrored.
// ---------------------------------------------------------------------------
__global__ void __launch_bounds__(256)
kde_wmma(const float4* __restrict__ A, const float4* __restrict__ B,
         float* __restrict__ out, int ns, int split) {
    const int wave  = blockIdx.x * (blockDim.x >> 5) + (threadIdx.x >> 5);
    const int lane  = threadIdx.x & 31;
    const int l16   = lane & 15;
    const bool hi   = lane >= 16;

    const int tileN   = wave / split;       // which 16-location tile
    const int chunk   = wave % split;       // which sample chunk
    const int chunk_m = ns / split;         // samples per chunk
    const int iters   = chunk_m >> 4;       // 16-sample steps
    const int m_base  = chunk * chunk_m;

    // B operand: constant across the loop
    float4 bl = B[tileN * 16 + l16];
    v2f bv;
    bv.x = hi ? bl.z : bl.x;
    bv.y = hi ? bl.w : bl.y;

    v8f acc = {};   // running sum of exp2(dot) over samples

    for (int it = 0; it < iters; ++it) {
        float4 as = A[m_base + it * 16 + l16];
        v2f av;
        av.x = hi ? as.z : as.x;
        av.y = hi ? as.w : as.y;

        v8f c = {};
        // D = A(16x4) x B(4x16) + 0  ->  16x16 tile of -|s-l|^2 * log2e / BW^2
        v8f d = __builtin_amdgcn_wmma_f32_16x16x4_f32(
            /*neg_a=*/false, av, /*neg_b=*/false, bv,
            /*c_mod=*/(short)0, c, /*reuse_a=*/false, /*reuse_b=*/false);

#pragma unroll
        for (int k = 0; k < 8; ++k)
            acc[k] += __builtin_amdgcn_exp2f(d[k]);
    }

    // Reduce over M: 8 accumulator VGPRs cover M=v (lanes<16) / M=v+8 (lanes>=16)
    float s = acc[0] + acc[1] + acc[2] + acc[3] + acc[4] + acc[5] + acc[6] + acc[7];
    s += __shfl_xor(s, 16, 32);     // combine the two lane halves (same N)

    if (lane < 16) atomicAdd(&out[tileN * 16 + l16], s);
}

// ---------------------------------------------------------------------------
// Kernel 5: norm = sum(out); out /= norm  (single block)
// ---------------------------------------------------------------------------
__global__ void kde_normalize(float* __restrict__ out, int nl) {
    __shared__ float red[256];
    __shared__ float norm;
    float s = 0.f;
    for (int i = threadIdx.x; i < nl; i += blockDim.x) s += out[i];
    red[threadIdx.x] = s;
    __syncthreads();
    for (int st = blockDim.x >> 1; st > 0; st >>= 1) {
        if ((int)threadIdx.x < st) red[threadIdx.x] += red[threadIdx.x + st];
        __syncthreads();
    }
    if (threadIdx.x == 0) norm = red[0];
    __syncthreads();
    float inv = 1.0f / norm;
    for (int i = threadIdx.x; i < nl; i += blockDim.x) out[i] *= inv;
}

// ---------------------------------------------------------------------------
extern "C" void kernel_launch(void* const* d_in, const int* in_sizes, int n_in,
                              void* d_out, int out_size, void* d_ws, size_t ws_size,
                              hipStream_t stream) {
    const float* samples   = (const float*)d_in[0];
    const float* locations = (const float*)d_in[1];
    const int ns = in_sizes[0] / 2;   // 16384
    const int nl = in_sizes[1] / 2;   // 8192
    float* out = (float*)d_out;

    // Workspace layout: [absmax(2f, padded 256B)] [A: ns float4] [B: nl float4]
    char* ws = (char*)d_ws;
    float*  absmax = (float*)ws;
    float4* A      = (float4*)(ws + 256);
    float4* B      = (float4*)(ws + 256 + (size_t)ns * sizeof(float4));

    kde_absmax<<<1, 256, 0, stream>>>(locations, absmax, nl);
    kde_prep_samples<<<(ns + 255) / 256, 256, 0, stream>>>(samples, absmax, A, ns);
    kde_prep_locations<<<(nl + 255) / 256, 256, 0, stream>>>(locations, B, out, nl);

    // 512 location tiles x 16 sample chunks = 8192 waves; 8 waves per block.
    const int split  = 16;
    const int waves  = (nl / 16) * split;
    kde_wmma<<<waves / 8, 256, 0, stream>>>(A, B, out, ns, split);

    kde_normalize<<<1, 256, 0, stream>>>(out, nl);
}